// Decodering_model_33406255629187
// MI455X (gfx1250) — compile-verified
//
#include <hip/hip_runtime.h>
#include <math.h>

#define Mrows 972
#define Ncols 1944
#define BATCH 8
#define DEG   8
#define NEDGE (Mrows * DEG)   /* 7776  */
#define NITER 5
#define CMAX  24              /* max column degree we record (E[deg]=4) */

// ---------------------------------------------------------------------------
// Tensor Data Mover: stage a contiguous [dim1][dim0] f32 tile into LDS.
// D# packing per CDNA5 ISA §8 (group0: count/lds_addr/global_addr/type=2,
// group1: data_size=4B, tensor dims == tile dims, stride = dim0).
// clang-23 builtin arity: (u32x4, i32x8, i32x4, i32x4, i32x8, i32 cpol).
// ---------------------------------------------------------------------------
typedef __attribute__((ext_vector_type(4))) unsigned tdm_u4;
typedef __attribute__((ext_vector_type(8))) int      tdm_i8;
typedef __attribute__((ext_vector_type(4))) int      tdm_i4;

__device__ __forceinline__ void tdm_load_f32_2d(const float* gptr, void* lds_ptr,
                                                unsigned dim0, unsigned dim1)
{
    unsigned long long ga = (unsigned long long)(uintptr_t)gptr;
    unsigned lds_off = (unsigned)(uintptr_t)lds_ptr; // low 32 bits = LDS byte offset

    tdm_u4 g0;
    g0.x = 1u;                                         // count=1 (valid D#)
    g0.y = lds_off;                                    // lds_addr [63:32]
    g0.z = (unsigned)(ga & 0xFFFFFFFFull);             // global_addr [95:64]
    g0.w = (unsigned)((ga >> 32) & 0x01FFFFFFull)      // global_addr [120:96]
         | (2u << 30);                                 // type=2 ("image")

    tdm_i8 g1;
    g1[0] = (int)(2u << 16);                           // data_size=2 -> 4 bytes
    g1[1] = (int)((dim0 & 0xFFFFu) << 16);             // tensor_dim0[15:0]
    g1[2] = (int)((dim0 >> 16) | ((dim1 & 0xFFFFu) << 16)); // dim0 hi | tensor_dim1 lo
    g1[3] = (int)((dim1 >> 16) | ((dim0 & 0xFFFFu) << 16)); // dim1 hi | tile_dim0
    g1[4] = (int)(dim1 & 0xFFFFu);                     // tile_dim1 (tile_dim2=0)
    g1[5] = (int)dim0;                                 // tensor_dim0_stride[31:0]
    g1[6] = 0;                                         // stride hi | dim1_stride lo
    g1[7] = 0;                                         // dim1_stride hi (unused, 2D)

    tdm_i4 gz4 = (tdm_i4)0;                            // groups 2/3 unused (<=2D)
    tdm_i8 gz8 = (tdm_i8)0;
    __builtin_amdgcn_tensor_load_to_lds(g0, g1, gz4, gz4, gz8, 0);
}

__device__ __forceinline__ float sgnf(float v)
{   // matches jnp.sign (sign(0)=0)
    return (v > 0.0f) ? 1.0f : ((v < 0.0f) ? -1.0f : 0.0f);
}

// ---------------------------------------------------------------------------
// One-time graph extraction from dense H.
// ---------------------------------------------------------------------------
__global__ __launch_bounds__(256)
void extract_rows(const float* __restrict__ H, int* __restrict__ cols /*[DEG][M]*/)
{
    int m = blockIdx.x * blockDim.x + threadIdx.x;
    if (m >= Mrows) return;
    int cnt = 0;
    for (int n = 0; n < Ncols; ++n) {
        if (H[(long)m * Ncols + n] != 0.0f && cnt < DEG)
            cols[cnt++ * Mrows + m] = n;
    }
    for (; cnt < DEG; ++cnt) cols[cnt * Mrows + m] = -1;
}

__global__ __launch_bounds__(256)
void extract_cols(const int* __restrict__ cols, int* __restrict__ cdeg,
                  int* __restrict__ crows /*[N][CMAX], stores i = e*M + m*/)
{
    int n = blockIdx.x * blockDim.x + threadIdx.x;
    if (n >= Ncols) return;
    int d = 0;
    for (int i = 0; i < NEDGE; ++i)           // deterministic order
        if (cols[i] == n && d < CMAX) crows[n * CMAX + d++] = i;
    cdeg[n] = d;
}

__global__ __launch_bounds__(256)
void zero_cv(float* __restrict__ cv)
{
    int i = blockIdx.x * blockDim.x + threadIdx.x;
    if (i < BATCH * NEDGE) cv[i] = 0.0f;
}

// ---------------------------------------------------------------------------
// Variable-node: total[b][n] = soft[b][n] + sum of cv over edges at column n.
// cv layout: [BATCH][DEG][M]; per-batch slice (31 KB) TDM-staged into LDS so
// the irregular CSC gather hits LDS, not VMEM.
// ---------------------------------------------------------------------------
__global__ __launch_bounds__(256)
void vn_kernel(const float* __restrict__ soft, const float* __restrict__ cv,
               const int* __restrict__ cdeg, const int* __restrict__ crows,
               float* __restrict__ dst)
{
    __shared__ float lcv[NEDGE];              // 31104 B
    const int b = blockIdx.y;
    if (threadIdx.x < 32) {                   // one TDM issue per block (wave 0)
        tdm_load_f32_2d(cv + (long)b * NEDGE, lcv, Mrows, DEG);
        __builtin_amdgcn_s_wait_tensorcnt(0);
    }
    __syncthreads();

    int n = blockIdx.x * blockDim.x + threadIdx.x;
    if (n >= Ncols) return;
    float t = soft[b * Ncols + n];
    int d = cdeg[n];
    for (int k = 0; k < d; ++k)
        t += lcv[crows[n * CMAX + k]];
    dst[b * Ncols + n] = t;
}

// ---------------------------------------------------------------------------
// Check-node: normalized min-sum update per (b, m). The full total[8][1944]
// table (62 KB) is TDM-staged into LDS; per-edge column gathers hit LDS.
// ---------------------------------------------------------------------------
__global__ __launch_bounds__(256)
void check_kernel(const float* __restrict__ total, const int* __restrict__ cols,
                  float* __restrict__ cv, const float* __restrict__ cwp)
{
    __shared__ float lt[BATCH * Ncols];       // 62208 B
    if (threadIdx.x < 32) {
        tdm_load_f32_2d(total, lt, Ncols, BATCH);
        __builtin_amdgcn_s_wait_tensorcnt(0);
    }
    __syncthreads();

    int tid = blockIdx.x * blockDim.x + threadIdx.x;
    if (tid >= BATCH * Mrows) return;
    const int b = tid / Mrows;
    const int m = tid - b * Mrows;

    float x = cwp[0];
    float w = (x > 20.0f) ? x : log1pf(expf(x));   // softplus(check_weight)

    float vcv[DEG], av[DEG];
    int   cix[DEG];
    float min1 = 3.0e38f, min2 = 3.0e38f, prod = 1.0f;
    float* cvb = cv + (long)b * NEDGE;             // [DEG][M] slice

    #pragma unroll
    for (int e = 0; e < DEG; ++e) {
        int c = cols[e * Mrows + m];
        cix[e] = c;
        float old = (c >= 0) ? cvb[e * Mrows + m] : 0.0f;
        float t   = (c >= 0) ? lt[b * Ncols + c]  : 0.0f;
        float v   = t - old;                       // extrinsic v->c message
        vcv[e] = v;
        float a = fminf(fabsf(v), 1.0e30f);        // clip like reference
        if (c < 0) a = 3.0e38f;                    // non-edge never wins min
        av[e] = a;
        if (a < min1) { min2 = min1; min1 = a; }
        else if (a < min2) { min2 = a; }
        if (c >= 0) prod *= sgnf(v);               // sign(0)=0 zeroes the row
    }

    #pragma unroll
    for (int e = 0; e < DEG; ++e) {
        if (cix[e] < 0) continue;
        float s    = sgnf(vcv[e]);
        float emin = (av[e] > min1) ? min1 : min2; // exclude-self min
        cvb[e * Mrows + m] = emin * (prod * s) * w;
    }
}

// ---------------------------------------------------------------------------
extern "C" void kernel_launch(void* const* d_in, const int* in_sizes, int n_in,
                              void* d_out, int out_size, void* d_ws, size_t ws_size,
                              hipStream_t stream)
{
    const float* soft = (const float*)d_in[0];    // [8][1944]
    const float* H    = (const float*)d_in[1];    // [972][1944]
    const float* cw   = (const float*)d_in[2];    // [1]
    float*       out  = (float*)d_out;            // [8][1944]

    char* ws = (char*)d_ws;
    int*   cols  = (int*)(ws);                               //  31104 B
    int*   cdeg  = (int*)(ws + 31104);                       //   7776 B
    int*   crows = (int*)(ws + 38880);                       // 186624 B
    float* cv    = (float*)(ws + 225504);                    // 248832 B
    float* total = (float*)(ws + 474336);                    //  62208 B  (~537 KB)

    extract_rows<<<(Mrows + 255) / 256, 256, 0, stream>>>(H, cols);
    extract_cols<<<(Ncols + 255) / 256, 256, 0, stream>>>(cols, cdeg, crows);
    zero_cv<<<(BATCH * NEDGE + 255) / 256, 256, 0, stream>>>(cv);

    dim3 vgrid((Ncols + 255) / 256, BATCH);
    for (int it = 0; it < NITER; ++it) {
        vn_kernel<<<vgrid, 256, 0, stream>>>(soft, cv, cdeg, crows, total);
        check_kernel<<<(BATCH * Mrows + 255) / 256, 256, 0, stream>>>(total, cols, cv, cw);
    }
    // marginalize straight into d_out
    vn_kernel<<<vgrid, 256, 0, stream>>>(soft, cv, cdeg, crows, out);
}